// MeshPotential_79482664780400
// MI455X (gfx1250) — compile-verified
//
#include <hip/hip_runtime.h>
#include <hip/hip_bf16.h>
#include <math.h>

// ---------------------------------------------------------------------------
// P3M mesh potential on gfx1250 (MI455X).
//   scatter (atomic f32)  ->  3x fwd DFT pass (WMMA f32 16x16x4)  ->  G(k)
//   ->  3x inv DFT pass   ->  gather + self-term subtraction.
// Complex 128^3 grid = 16MB: L2-resident, so DFT == GEMM on matrix pipes.
// Forward/inverse conjugation is baked into dual (+sin/-sin) twiddle tables
// so the WMMA inner loop carries zero sign-fixup VALU work.
// ---------------------------------------------------------------------------

typedef __attribute__((ext_vector_type(2))) float v2f;
typedef __attribute__((ext_vector_type(8))) float v8f;

#define NXYZ 128
#define NM   (128 * 128 * 128)      // 2,097,152 grid points
#define QTOT (128 * 128)            // 16384 lines per pass
#define TWSZ (128 * 128)            // twiddle table entries
#define SIGMA 1.0f
#define SELF_COEF 0.7978845608028654f   // sqrt(2/pi)/sigma
#define TWO_PI 6.283185307179586f

// ---- 3x3 inverse + det (cell rows = basis vectors, row-major) -------------
__device__ inline void invert3x3(const float* __restrict__ c, float inv[9], float& det) {
    float a00 = c[0], a01 = c[1], a02 = c[2];
    float a10 = c[3], a11 = c[4], a12 = c[5];
    float a20 = c[6], a21 = c[7], a22 = c[8];
    det = a00 * (a11 * a22 - a12 * a21)
        + a01 * (a12 * a20 - a10 * a22)
        + a02 * (a10 * a21 - a11 * a20);
    float id = 1.0f / det;
    inv[0] = (a11 * a22 - a12 * a21) * id;
    inv[1] = (a02 * a21 - a01 * a22) * id;
    inv[2] = (a01 * a12 - a02 * a11) * id;
    inv[3] = (a12 * a20 - a10 * a22) * id;
    inv[4] = (a00 * a22 - a02 * a20) * id;
    inv[5] = (a02 * a10 - a00 * a12) * id;
    inv[6] = (a10 * a21 - a11 * a20) * id;
    inv[7] = (a01 * a20 - a00 * a21) * id;
    inv[8] = (a00 * a11 - a01 * a10) * id;
}

// ---- order-4 P3M assignment weights ---------------------------------------
__device__ inline void p3m_w4(float x, float w[4]) {
    float x2 = x * x, x3 = x2 * x;
    const float s = 1.0f / 48.0f;
    w[0] = (1.0f  - 6.0f  * x + 12.0f * x2 - 8.0f  * x3) * s;
    w[1] = (23.0f - 30.0f * x - 12.0f * x2 + 24.0f * x3) * s;
    w[2] = (23.0f + 30.0f * x - 12.0f * x2 - 24.0f * x3) * s;
    w[3] = (1.0f  + 6.0f  * x + 12.0f * x2 + 8.0f  * x3) * s;
}

__device__ inline void atom_stencil(const float inv[9], float p0, float p1, float p2,
                                    int idx[3][4], float w[3][4]) {
    #pragma unroll
    for (int j = 0; j < 3; ++j) {
        float rel = (p0 * inv[0 * 3 + j] + p1 * inv[1 * 3 + j] + p2 * inv[2 * 3 + j]) * 128.0f;
        float f = floorf(rel);
        float x = rel - (f + 0.5f);
        p3m_w4(x, w[j]);
        int i0 = (int)f;
        #pragma unroll
        for (int s = 0; s < 4; ++s)
            idx[j][s] = (i0 + s - 1 + 256) & 127;
    }
}

// ---- zero fill -------------------------------------------------------------
__global__ void zero_kernel(float* __restrict__ p, int n) {
    int i = blockIdx.x * blockDim.x + threadIdx.x;
    if (i < n) p[i] = 0.0f;
}

// ---- twiddle tables: F[k][n] = exp(-2*pi*i*k*n/128) -----------------------
// cosT = Re(F); sinP = Im(F) (forward); sinN = -Im(F).
// Forward pass uses (Ai = sinP, AiN = sinN); inverse swaps the two pointers.
__global__ void twiddle_kernel(float* __restrict__ cosT,
                               float* __restrict__ sinP,
                               float* __restrict__ sinN) {
    int i = blockIdx.x * blockDim.x + threadIdx.x;   // 16384
    if (i >= TWSZ) return;
    int k = i >> 7, n = i & 127;
    int p = (k * n) & 127;                            // periodic
    float ang = -TWO_PI * (float)p * (1.0f / 128.0f);
    float c = cosf(ang), s = sinf(ang);
    cosT[i] = c;
    sinP[i] = s;
    sinN[i] = -s;
}

// ---- charge scatter (points_to_mesh) --------------------------------------
__global__ void scatter_kernel(const float* __restrict__ cell,
                               const float* __restrict__ pos,
                               const float* __restrict__ chg,
                               float* __restrict__ rho, int natoms) {
    int i = blockIdx.x * blockDim.x + threadIdx.x;
    if (i >= natoms) return;
    float inv[9], det;
    invert3x3(cell, inv, det);
    int idx[3][4]; float w[3][4];
    atom_stencil(inv, pos[3 * i], pos[3 * i + 1], pos[3 * i + 2], idx, w);
    float q = chg[i];
    #pragma unroll
    for (int a = 0; a < 4; ++a) {
        float qa = q * w[0][a];
        int ia = idx[0][a] * 128;
        #pragma unroll
        for (int b = 0; b < 4; ++b) {
            float qab = qa * w[1][b];
            int iab = (ia + idx[1][b]) * 128;
            #pragma unroll
            for (int c = 0; c < 4; ++c)
                unsafeAtomicAdd(&rho[iab + idx[2][c]], qab * w[2][c]);
        }
    }
}

// ---- one DFT pass along the slowest axis, via V_WMMA_F32_16X16X4_F32 ------
// Logical input layout: in[n * 16384 + q], n = transform axis (slowest),
// q = contiguous batch. Output: out[q * 128 + k] -> transformed axis becomes
// fastest; three passes rotate the layout back to the original order.
__global__ __launch_bounds__(256) void dft_pass_kernel(
        const float* __restrict__ inRe, const float* __restrict__ inIm,
        float* __restrict__ outRe, float* __restrict__ outIm,
        const float* __restrict__ cosT,
        const float* __restrict__ sinA,     // Im(F) for this direction
        const float* __restrict__ sinAN) {  // -Im(F) for this direction
    __shared__ float ldsRe[128 * 16];
    __shared__ float ldsIm[128 * 16];

    const int t  = threadIdx.x;
    const int q0 = blockIdx.x * 16;

    // Stage the 16 input lines (128 x 16 Re + Im) into LDS; 8 waves reuse them.
    #pragma unroll
    for (int i = 0; i < 8; ++i) {
        int idx = t + i * 256;              // 0..2047
        int n = idx >> 4, q = idx & 15;
        ldsRe[idx] = inRe[n * QTOT + q0 + q];
        ldsIm[idx] = inIm[n * QTOT + q0 + q];
    }
    __syncthreads();

    const int lane = t & 31;
    const int wave = t >> 5;
    const int k0   = wave * 16;             // 16-frequency tile per wave
    const int m    = lane & 15;             // A-row (freq) / B-col (batch line)
    const int j2   = (lane >> 4) * 2;       // K sub-offset for this half-wave

    v8f accR = {};
    v8f accI = {};
    const float* twc = cosT  + (k0 + m) * 128;
    const float* twp = sinA  + (k0 + m) * 128;
    const float* twn = sinAN + (k0 + m) * 128;

    for (int nr = 0; nr < 128; nr += 4) {
        int ka = nr + j2;
        v2f Ar  = { twc[ka], twc[ka + 1] }; // Re(F)
        v2f Ai  = { twp[ka], twp[ka + 1] }; // Im(F)
        v2f AiN = { twn[ka], twn[ka + 1] }; // -Im(F)
        v2f Br = { ldsRe[ka * 16 + m], ldsRe[(ka + 1) * 16 + m] };
        v2f Bi = { ldsIm[ka * 16 + m], ldsIm[(ka + 1) * 16 + m] };
        // Yr += Fr*Xr - Fi*Xi ; Yi += Fr*Xi + Fi*Xr
        accR = __builtin_amdgcn_wmma_f32_16x16x4_f32(false, Ar,  false, Br, (short)0, accR, false, false);
        accR = __builtin_amdgcn_wmma_f32_16x16x4_f32(false, AiN, false, Bi, (short)0, accR, false, false);
        accI = __builtin_amdgcn_wmma_f32_16x16x4_f32(false, Ar,  false, Bi, (short)0, accI, false, false);
        accI = __builtin_amdgcn_wmma_f32_16x16x4_f32(false, Ai,  false, Br, (short)0, accI, false, false);
    }

    // C/D layout: VGPR r, lanes 0-15 -> M=r, lanes 16-31 -> M=r+8; N = lane&15
    const int N  = m;
    const int mh = (lane >> 4) * 8;
    #pragma unroll
    for (int r = 0; r < 8; ++r) {
        int addr = (q0 + N) * 128 + (k0 + r + mh);
        outRe[addr] = accR[r];
        outIm[addr] = accI[r];
    }
}

// ---- Fourier-space Green's function multiply ------------------------------
__global__ void gmul_kernel(const float* __restrict__ cell,
                            float* __restrict__ re, float* __restrict__ im) {
    int i = blockIdx.x * blockDim.x + threadIdx.x;
    if (i >= NM) return;
    float inv[9], det;
    invert3x3(cell, inv, det);
    int kx = i >> 14, ky = (i >> 7) & 127, kz = i & 127;
    float mx = (float)(kx < 64 ? kx : kx - 128);
    float my = (float)(ky < 64 ? ky : ky - 128);
    float mz = (float)(kz < 64 ? kz : kz - 128);
    float ksq = 0.0f;
    #pragma unroll
    for (int j = 0; j < 3; ++j) {
        float kj = TWO_PI * (mx * inv[j * 3 + 0] + my * inv[j * 3 + 1] + mz * inv[j * 3 + 2]);
        ksq += kj * kj;
    }
    float g = 0.0f;
    if (ksq > 0.0f)
        g = (4.0f * (float)M_PI / ksq) * expf(-0.5f * SIGMA * SIGMA * ksq) / fabsf(det);
    re[i] *= g;
    im[i] *= g;
}

// ---- potential gather (mesh_to_points) + self term ------------------------
__global__ void gather_kernel(const float* __restrict__ cell,
                              const float* __restrict__ pos,
                              const float* __restrict__ chg,
                              const float* __restrict__ pot,
                              float* __restrict__ out, int natoms) {
    int i = blockIdx.x * blockDim.x + threadIdx.x;
    if (i >= natoms) return;
    float inv[9], det;
    invert3x3(cell, inv, det);
    int idx[3][4]; float w[3][4];
    atom_stencil(inv, pos[3 * i], pos[3 * i + 1], pos[3 * i + 2], idx, w);
    float acc = 0.0f;
    #pragma unroll
    for (int a = 0; a < 4; ++a) {
        int ia = idx[0][a] * 128;
        #pragma unroll
        for (int b = 0; b < 4; ++b) {
            int iab = (ia + idx[1][b]) * 128;
            float wab = w[0][a] * w[1][b];
            #pragma unroll
            for (int c = 0; c < 4; ++c)
                acc += wab * w[2][c] * pot[iab + idx[2][c]];
        }
    }
    out[i] = acc - chg[i] * SELF_COEF;
}

// ---------------------------------------------------------------------------
extern "C" void kernel_launch(void* const* d_in, const int* in_sizes, int n_in,
                              void* d_out, int out_size, void* d_ws, size_t ws_size,
                              hipStream_t stream) {
    (void)n_in; (void)out_size; (void)ws_size;
    const float* cell = (const float*)d_in[0];
    const float* pos  = (const float*)d_in[1];
    const float* chg  = (const float*)d_in[2];
    float* out = (float*)d_out;
    const int natoms = in_sizes[2];          // charges: (N,1)

    float* ws   = (float*)d_ws;
    float* Re0  = ws;                        // 4 x 8MB ping-pong complex grids
    float* Im0  = ws + (size_t)NM;
    float* Re1  = ws + (size_t)2 * NM;
    float* Im1  = ws + (size_t)3 * NM;
    float* cosT = ws + (size_t)4 * NM;       // 3 x 64KB twiddle tables
    float* sinP = cosT + TWSZ;
    float* sinN = sinP + TWSZ;

    twiddle_kernel<<<64, 256, 0, stream>>>(cosT, sinP, sinN);
    zero_kernel<<<(2 * NM + 255) / 256, 256, 0, stream>>>(Re0, 2 * NM);  // Re0+Im0
    scatter_kernel<<<(natoms + 255) / 256, 256, 0, stream>>>(cell, pos, chg, Re0, natoms);

    // forward FFT: three layout-rotating passes (Ai = +ImF = sinP)
    dft_pass_kernel<<<QTOT / 16, 256, 0, stream>>>(Re0, Im0, Re1, Im1, cosT, sinP, sinN);
    dft_pass_kernel<<<QTOT / 16, 256, 0, stream>>>(Re1, Im1, Re0, Im0, cosT, sinP, sinN);
    dft_pass_kernel<<<QTOT / 16, 256, 0, stream>>>(Re0, Im0, Re1, Im1, cosT, sinP, sinN);

    gmul_kernel<<<(NM + 255) / 256, 256, 0, stream>>>(cell, Re1, Im1);

    // inverse FFT: conjugated twiddles (swap sin tables), unnormalized
    dft_pass_kernel<<<QTOT / 16, 256, 0, stream>>>(Re1, Im1, Re0, Im0, cosT, sinN, sinP);
    dft_pass_kernel<<<QTOT / 16, 256, 0, stream>>>(Re0, Im0, Re1, Im1, cosT, sinN, sinP);
    dft_pass_kernel<<<QTOT / 16, 256, 0, stream>>>(Re1, Im1, Re0, Im0, cosT, sinN, sinP);

    gather_kernel<<<(natoms + 255) / 256, 256, 0, stream>>>(cell, pos, chg, Re0, out, natoms);
}